// BOW_SVM_23029614641670
// MI455X (gfx1250) — compile-verified
//
#include <hip/hip_runtime.h>
#include <hip/hip_bf16.h>

typedef float v2f __attribute__((ext_vector_type(2)));
typedef float v8f __attribute__((ext_vector_type(8)));

#define B_   2048
#define S_   256
#define D_   768
#define C_   100

// ---------------------------------------------------------------------------
// Kernel 1: masked mean pooling.  One block per batch row, 256 threads.
// Each thread owns 3 columns (t, t+256, t+512). The mask test is wave-uniform
// (read from LDS, same value for every lane) so skipped tokens cost ~nothing.
// ---------------------------------------------------------------------------
__global__ void __launch_bounds__(256) bow_pool_kernel(
    const int* __restrict__ ids, const int* __restrict__ mask,
    const float* __restrict__ table, float* __restrict__ bow) {
  __shared__ int   s_ids[S_];
  __shared__ float s_msk[S_];
  const int b = blockIdx.x;
  const int t = threadIdx.x;
  s_ids[t] = ids[(size_t)b * S_ + t];
  s_msk[t] = (float)mask[(size_t)b * S_ + t];
  __syncthreads();

  float a0 = 0.f, a1 = 0.f, a2 = 0.f, cnt = 0.f;
  const int c0 = t, c1 = t + 256, c2 = t + 512;
  for (int s = 0; s < S_; ++s) {
    const float m = s_msk[s];
    cnt += m;
    if (m != 0.f) {                     // uniform across the block -> scalar branch
      const float* __restrict__ row = table + (size_t)s_ids[s] * D_;
      a0 += row[c0];
      a1 += row[c1];
      a2 += row[c2];
    }
  }
  const float inv = 1.0f / cnt;         // >=1 valid token guaranteed
  float* __restrict__ o = bow + (size_t)b * D_;
  o[c0] = a0 * inv;
  o[c1] = a1 * inv;
  o[c2] = a2 * inv;
}

// ---------------------------------------------------------------------------
// Kernel 2: H = relu(bow @ W1 + b1), M=2048 N=768 K=768, f32 WMMA 16x16x4.
// 8 waves/block; wave w computes rows [.. +16) x cols [.. +64) => 4 accs,
// one shared A-fragment per K-step (4 WMMAs reuse it).
// grid = (2048/128, 768/64) = (16, 12)
// ---------------------------------------------------------------------------
__global__ void __launch_bounds__(256) gemm1_relu_kernel(
    const float* __restrict__ A, const float* __restrict__ W1,
    const float* __restrict__ b1, float* __restrict__ H) {
  const int wave = threadIdx.x >> 5;
  const int lane = threadIdx.x & 31;
  const int tileM = (blockIdx.x * 8 + wave) * 16;
  const int tileN = blockIdx.y * 64;
  const int mrow  = lane & 15;
  const int khalf = (lane >> 4) * 2;     // lanes 0-15 -> K+0/1, lanes 16-31 -> K+2/3

  v8f acc[4] = {v8f{}, v8f{}, v8f{}, v8f{}};
  const float* __restrict__ arow = A + (size_t)(tileM + mrow) * D_;

  for (int k = 0; k < D_; k += 4) {
    const int kk = k + khalf;
    v2f af;
    af[0] = arow[kk];
    af[1] = arow[kk + 1];
    const float* __restrict__ b0 = W1 + (size_t)kk * D_ + tileN + mrow;
    const float* __restrict__ b1p = b0 + D_;
#pragma unroll
    for (int nt = 0; nt < 4; ++nt) {
      v2f bf;
      bf[0] = b0[nt * 16];
      bf[1] = b1p[nt * 16];
      acc[nt] = __builtin_amdgcn_wmma_f32_16x16x4_f32(
          false, af, false, bf, (short)0, acc[nt], false, false);
    }
  }

  const int rowoff = (lane >> 4) * 8;    // C/D layout: VGPR i -> M=i / M=i+8
#pragma unroll
  for (int nt = 0; nt < 4; ++nt) {
    const int col  = tileN + nt * 16 + mrow;
    const float bb = b1[col];
#pragma unroll
    for (int i = 0; i < 8; ++i) {
      const int row = tileM + rowoff + i;
      float v = acc[nt][i] + bb;
      H[(size_t)row * D_ + col] = v > 0.f ? v : 0.f;
    }
  }
}

// ---------------------------------------------------------------------------
// Kernel 3: scores = H @ W2 + b2, M=2048 N=100(->7 tiles of 16) K=768.
// One 16x16 tile per wave; B column clamped so padded-column loads stay in
// bounds (they feed only columns that are never stored). Store guard comes
// AFTER all WMMAs so EXEC is all-ones for the matrix ops.
// grid = (2048/128, 7)
// ---------------------------------------------------------------------------
__global__ void __launch_bounds__(256) gemm2_kernel(
    const float* __restrict__ H, const float* __restrict__ W2,
    const float* __restrict__ b2, float* __restrict__ out) {
  const int wave = threadIdx.x >> 5;
  const int lane = threadIdx.x & 31;
  const int tileM = (blockIdx.x * 8 + wave) * 16;
  const int tileN = blockIdx.y * 16;
  const int mrow  = lane & 15;
  const int khalf = (lane >> 4) * 2;
  const int col   = tileN + mrow;
  const int colc  = col < C_ ? col : (C_ - 1);

  v8f acc = v8f{};
  const float* __restrict__ arow = H + (size_t)(tileM + mrow) * D_;

  for (int k = 0; k < D_; k += 4) {
    const int kk = k + khalf;
    v2f af;
    af[0] = arow[kk];
    af[1] = arow[kk + 1];
    v2f bf;
    bf[0] = W2[(size_t)kk * C_ + colc];
    bf[1] = W2[(size_t)(kk + 1) * C_ + colc];
    acc = __builtin_amdgcn_wmma_f32_16x16x4_f32(
        false, af, false, bf, (short)0, acc, false, false);
  }

  if (col < C_) {
    const float bb = b2[col];
    const int rowoff = (lane >> 4) * 8;
#pragma unroll
    for (int i = 0; i < 8; ++i) {
      const int row = tileM + rowoff + i;
      const float v = acc[i] + bb;
      out[(size_t)row * C_ + col] = v;                    // tuple elem 0
      out[(size_t)B_ * C_ + (size_t)row * C_ + col] = v;  // tuple elem 1
    }
  }
}

// ---------------------------------------------------------------------------
extern "C" void kernel_launch(void* const* d_in, const int* in_sizes, int n_in,
                              void* d_out, int out_size, void* d_ws, size_t ws_size,
                              hipStream_t stream) {
  const int*   ids   = (const int*)d_in[0];
  const int*   mask  = (const int*)d_in[1];
  const float* table = (const float*)d_in[2];
  const float* W1    = (const float*)d_in[3];
  const float* b1    = (const float*)d_in[4];
  const float* W2    = (const float*)d_in[5];
  const float* b2    = (const float*)d_in[6];
  float*       out   = (float*)d_out;

  float* bow = (float*)d_ws;                     // [2048 x 768] = 6 MB
  float* h   = bow + (size_t)B_ * D_;            // [2048 x 768] = 6 MB

  bow_pool_kernel<<<B_, 256, 0, stream>>>(ids, mask, table, bow);
  gemm1_relu_kernel<<<dim3(B_ / 128, D_ / 64), 256, 0, stream>>>(bow, W1, b1, h);
  gemm2_kernel<<<dim3(B_ / 128, 7), 256, 0, stream>>>(h, W2, b2, out);
}